// TraversabilityFilter_16535624090384
// MI455X (gfx1250) — compile-verified
//
#include <hip/hip_runtime.h>

typedef __attribute__((ext_vector_type(2))) float v2f;
typedef __attribute__((ext_vector_type(8))) float v8f;

#define HW      4096
#define OUTDIM  4090
#define TILE_W  64
#define TILE_H  64
#define IN_ROWS 70          // TILE_H + 6
#define IN_COLS 70          // TILE_W + 6
#define PITCH   72          // padded LDS pitch (floats)

#if defined(__HIP_DEVICE_COMPILE__) && defined(__has_builtin)
#if __has_builtin(__builtin_amdgcn_global_load_async_to_lds_b64) && \
    __has_builtin(__builtin_amdgcn_s_wait_asynccnt)
#define HAVE_ASYNC_LDS 1
#endif
#endif

#ifdef HAVE_ASYNC_LDS
typedef int v2i_vs __attribute__((vector_size(8)));
typedef __attribute__((address_space(1))) v2i_vs* async_gptr;
typedef __attribute__((address_space(3))) v2i_vs* async_lptr;
#endif

// Tap k -> (dy,dx) offset inside the 7x7 input window (center = 3,3).
// k 0..8  : dilation 1 (channels 0-3)
// k 9..17 : dilation 2 (channels 4-7)
// k 18..26: dilation 3 (channels 8-11)
// k 27    : zero-weight pad (points at center; value irrelevant since A=0)
__device__ __forceinline__ void tap_pos(int k, int& dy, int& dx) {
    if (k >= 27) { dy = 3; dx = 3; return; }
    int d = k / 9 + 1;            // dilation
    int t = k - (d - 1) * 9;
    int a = t / 3, b = t - a * 3; // 3x3 kernel coords
    dy = 3 + d * (a - 1);
    dx = 3 + d * (b - 1);
}

// Weight matrix W[16][28]: channel c, tap k (zero outside each conv's block).
__device__ __forceinline__ float wget(const float* w1, const float* w2,
                                      const float* w3, int c, int k) {
    if (k < 9)       return (c < 4)             ? w1[c * 9 + k]            : 0.f;
    else if (k < 18) return (c >= 4 && c < 8)   ? w2[(c - 4) * 9 + k - 9]  : 0.f;
    else if (k < 27) return (c >= 8 && c < 12)  ? w3[(c - 8) * 9 + k - 18] : 0.f;
    return 0.f;
}

__global__ __launch_bounds__(256) void trav_wmma_kernel(
    const float* __restrict__ elev,
    const float* __restrict__ w1,
    const float* __restrict__ w2,
    const float* __restrict__ w3,
    const float* __restrict__ wo,
    float* __restrict__ out)
{
    __shared__ float lds[IN_ROWS * PITCH];

    const int tid    = threadIdx.x;
    const int out_r0 = blockIdx.y * TILE_H;
    const int out_c0 = blockIdx.x * TILE_W;

    // ---- stage 70x70 input tile into LDS ----
    const bool interior = (out_r0 + IN_ROWS <= HW) && (out_c0 + IN_COLS <= HW);
    if (interior) {
        // 70 = 35 x 2: whole tile as aligned B64 transfers, no remainder.
        for (int idx = tid; idx < IN_ROWS * (IN_COLS / 2); idx += 256) {
            int r  = idx / (IN_COLS / 2);
            int c2 = (idx - r * (IN_COLS / 2)) * 2;
            const float* gp = &elev[(out_r0 + r) * HW + out_c0 + c2];
            float*       lp = &lds[r * PITCH + c2];
#ifdef HAVE_ASYNC_LDS
            __builtin_amdgcn_global_load_async_to_lds_b64(
                (async_gptr)(__attribute__((address_space(1))) void*)(void*)gp,
                (async_lptr)lp,
                0, 0);
#else
            *(float2*)lp = *(const float2*)gp;
#endif
        }
#ifdef HAVE_ASYNC_LDS
        __builtin_amdgcn_s_wait_asynccnt(0);
#endif
    } else {
        // edge tiles: scalar, clamped
        for (int idx = tid; idx < IN_ROWS * IN_COLS; idx += 256) {
            int r  = idx / IN_COLS;
            int c  = idx - r * IN_COLS;
            int gr = min(out_r0 + r, HW - 1);
            int gc = min(out_c0 + c, HW - 1);
            lds[r * PITCH + c] = elev[gr * HW + gc];
        }
    }

    const int lane    = tid & 31;
    const int wave    = tid >> 5;
    const int n       = lane & 15;   // pixel index within strip (B/C column)
    const int half    = lane >> 4;   // 0: K=4g+0/1, ch 0-7 ; 1: K=4g+2/3, ch 8-15
    const int strip_c = (wave & 3) * 16;
    const int strip_r = (wave >> 2) * 32;

    // ---- per-lane constant A fragments (weights) and B LDS offsets ----
    v2f A[7];
    int off0[7], off1[7];
    const int ch = n;                // A-matrix row M = lane % 16 = channel
    #pragma unroll
    for (int g = 0; g < 7; ++g) {
        int k0 = 4 * g + 2 * half;
        int k1 = k0 + 1;
        A[g].x = wget(w1, w2, w3, ch, k0);
        A[g].y = wget(w1, w2, w3, ch, k1);
        int dy, dx;
        tap_pos(k0, dy, dx);
        off0[g] = dy * PITCH + strip_c + n + dx;
        tap_pos(k1, dy, dx);
        off1[g] = dy * PITCH + strip_c + n + dx;
    }

    // ---- per-lane epilogue weights (C layout: lanes 0-15 ch 0-7, 16-31 ch 8-15)
    float e[8];
    #pragma unroll
    for (int v = 0; v < 8; ++v)
        e[v] = (half == 0) ? wo[v] : ((v < 4) ? wo[8 + v] : 0.f);

    __syncthreads();

    const int gcol = out_c0 + strip_c + n;

    #pragma unroll 4
    for (int i = 0; i < 32; ++i) {
        const int R = strip_r + i;                 // tile-local output row
        const float* rb = &lds[R * PITCH];

        v8f acc = {0.f, 0.f, 0.f, 0.f, 0.f, 0.f, 0.f, 0.f};
        #pragma unroll
        for (int g = 0; g < 7; ++g) {
            v2f B;
            B.x = rb[off0[g]];
            B.y = rb[off1[g]];
            // D = A(16x4 weights) * B(4x16 pixels) + C, f32 throughout
            acc = __builtin_amdgcn_wmma_f32_16x16x4_f32(
                false, A[g], false, B, (short)0, acc, false, false);
        }

        // y = sum_c w_out[c] * |feat_c| ; channels split across lane halves
        float s = 0.f;
        #pragma unroll
        for (int v = 0; v < 8; ++v)
            s = fmaf(e[v], fabsf(acc[v]), s);
        float y = s + __shfl_xor(s, 16, 32);       // combine ch 0-7 with ch 8-11
        float r = __expf(-y);

        const int grow = out_r0 + R;
        if (half == 0 && grow < OUTDIM && gcol < OUTDIM)
            out[grow * OUTDIM + gcol] = r;
    }
}

extern "C" void kernel_launch(void* const* d_in, const int* in_sizes, int n_in,
                              void* d_out, int out_size, void* d_ws, size_t ws_size,
                              hipStream_t stream) {
    const float* elev = (const float*)d_in[0];
    const float* w1   = (const float*)d_in[1];
    const float* w2   = (const float*)d_in[2];
    const float* w3   = (const float*)d_in[3];
    const float* wo   = (const float*)d_in[4];
    float* out        = (float*)d_out;

    dim3 grid((OUTDIM + TILE_W - 1) / TILE_W,   // 64
              (OUTDIM + TILE_H - 1) / TILE_H);  // 64
    trav_wmma_kernel<<<grid, 256, 0, stream>>>(elev, w1, w2, w3, wo, out);
}